// MultiHeadPooling_74010876445112
// MI455X (gfx1250) — compile-verified
//
#include <hip/hip_runtime.h>

typedef float        v2f __attribute__((ext_vector_type(2)));
typedef float        v8f __attribute__((ext_vector_type(8)));
typedef unsigned int v4u __attribute__((ext_vector_type(4)));
typedef int          v4i __attribute__((ext_vector_type(4)));
typedef int          v8i __attribute__((ext_vector_type(8)));

namespace {
constexpr int   NB   = 32;
constexpr int   NS   = 4096;
constexpr int   ND   = 512;
constexpr float BIGV = 10000.0f;

constexpr int WAVES          = 4;                    // waves / workgroup (128 thr)
constexpr int TROWS          = 16;                   // rows per WMMA tile
constexpr int SCHUNK         = 256;                  // rows per workgroup
constexpr int NCHUNK         = NS / SCHUNK;          // 16
constexpr int ROWS_PER_WAVE  = SCHUNK / WAVES;       // 64
constexpr int TILES_PER_WAVE = ROWS_PER_WAVE / TROWS;// 4

// dynamic-LDS layout (byte offsets); 2 tile buffers per wave (double-buffered TDM)
constexpr unsigned LDS_W_OFF    = 0;                              // 512 f32 (w)
constexpr unsigned LDS_TILE_OFF = 2048;
constexpr unsigned TILE_BYTES   = TROWS * ND * 4;                 // 32 KiB
constexpr unsigned LDS_COMB_OFF = LDS_TILE_OFF + WAVES * 2 * TILE_BYTES; // 264192
constexpr int      COMB_FLOATS  = 2056;                           // 4*512 + m,l,cnt + pad
constexpr unsigned LDS_BYTES    = LDS_COMB_OFF + WAVES * COMB_FLOATS * 4; // ~290 KiB < 320 KiB
}

// Issue one TDM descriptor: DMA a 16x512 f32 tile (rows contiguous, stride ND)
// from global memory into LDS at byte offset lds_off.  EXEC-independent,
// completes in-order per wave, tracked by TENSORcnt.
__device__ __forceinline__ void tdm_load_tile(const float* gptr, unsigned lds_off)
{
#if __has_builtin(__builtin_amdgcn_tensor_load_to_lds)
    const unsigned long long ga = (unsigned long long)(uintptr_t)gptr;
    v4u g0;
    g0[0] = 1u;                                   // count=1 (valid user D#)
    g0[1] = lds_off;                              // lds_addr (bytes)
    g0[2] = (unsigned)(ga & 0xffffffffull);       // global_addr[31:0]
    g0[3] = (unsigned)((ga >> 32) & 0x01ffffffull) | 0x80000000u; // addr[56:32] | type=2
    v8i g1;
    g1[0] = (int)(2u << 16);                      // data_size = 4 bytes
    g1[1] = (int)((unsigned)ND << 16);            // tensor_dim0[15:0] = 512
    g1[2] = (int)(((unsigned)ND >> 16) | ((unsigned)TROWS << 16)); // dim0 hi | dim1 lo
    g1[3] = (int)((unsigned)ND << 16);            // tile_dim0 = 512
    g1[4] = TROWS;                                // tile_dim1 = 16, tile_dim2 = 0
    g1[5] = ND;                                   // tensor_dim0_stride = 512
    g1[6] = 0;
    g1[7] = 0;
    v4i z4 = {0, 0, 0, 0};
    v8i z8 = {0, 0, 0, 0, 0, 0, 0, 0};
    __builtin_amdgcn_tensor_load_to_lds(g0, g1, z4, z4, z8, 0);
#else
    (void)gptr; (void)lds_off;
#endif
}

// ---------------------------------------------------------------- pass 1 ----
extern "C" __global__ __launch_bounds__(WAVES * 32)
void mhp_pass1(const float* __restrict__ x, const unsigned char* __restrict__ mask,
               const float* __restrict__ w, float* __restrict__ partial)
{
    extern __shared__ float smem[];
    const int tid  = threadIdx.x;
    const int lane = tid & 31;
    const int wave = tid >> 5;
    const int b     = blockIdx.x / NCHUNK;
    const int chunk = blockIdx.x % NCHUNK;
    const int row0  = chunk * SCHUNK + wave * ROWS_PER_WAVE;
    const float* xbase = x + ((size_t)b * NS + row0) * ND;

    // stage w into LDS (2 KiB, shared by all waves)
    for (int i = tid; i < ND; i += blockDim.x) smem[LDS_W_OFF / 4 + i] = w[i];
    __syncthreads();

    const unsigned tile_lds0 = LDS_TILE_OFF + (unsigned)wave * 2u * TILE_BYTES;

    // per-lane accumulators: lane owns columns [lane*16, lane*16+16)
    const int c0 = lane * 16;
    float sumv[16], maxv[16], minv[16], accv[16];
#pragma unroll
    for (int i = 0; i < 16; ++i) { sumv[i] = 0.f; maxv[i] = -BIGV; minv[i] = BIGV; accv[i] = 0.f; }
    float m = -2.0f * BIGV, l = 0.f, cnt = 0.f;

#if __has_builtin(__builtin_amdgcn_tensor_load_to_lds)
    // prologue: start DMA of tile 0
    tdm_load_tile(xbase, tile_lds0);
#endif

    for (int t = 0; t < TILES_PER_WAVE; ++t) {
        const int srow = row0 + t * TROWS;
        const unsigned tile_lds = tile_lds0 + (unsigned)(t & 1) * TILE_BYTES;
        float* tile = &smem[tile_lds / 4];

#if __has_builtin(__builtin_amdgcn_tensor_load_to_lds)
        // double-buffer: kick tile t+1, then wait until tile t is resident.
        // TDM ops complete in-order per wave -> TENSORcnt<=1 means t is done.
        if (t + 1 < TILES_PER_WAVE) {
            tdm_load_tile(xbase + (size_t)(t + 1) * TROWS * ND,
                          tile_lds0 + (unsigned)((t + 1) & 1) * TILE_BYTES);
            __builtin_amdgcn_s_wait_tensorcnt(1);
        } else {
            __builtin_amdgcn_s_wait_tensorcnt(0);
        }
#else
        // fallback: cooperative wave copy (coalesced b128)
        for (int e = lane * 4; e < TROWS * ND; e += 32 * 4)
            *(float4*)&tile[e] = *(const float4*)&xbase[(size_t)t * TROWS * ND + e];
        __builtin_amdgcn_s_waitcnt(0);
#endif

        // ---- scores for 16 rows via v_wmma_f32_16x16x4_f32 -----------------
        // A: 16x4 tile fragment; B: w[k..k+3] replicated across N -> every
        // column of C holds that row's full dot product after 128 steps.
        const int khalf = (lane >> 4) << 1;  // lanes 0-15: K+0/1, lanes 16-31: K+2/3
        const int arow  = lane & 15;
        float sc[16];
#if __has_builtin(__builtin_amdgcn_wmma_f32_16x16x4_f32)
        v8f cf = {0.f, 0.f, 0.f, 0.f, 0.f, 0.f, 0.f, 0.f};
#pragma unroll 8
        for (int k = 0; k < ND; k += 4) {
            v2f af = *(const v2f*)&tile[arow * ND + k + khalf];
            v2f bf = *(const v2f*)&smem[LDS_W_OFF / 4 + k + khalf];
            cf = __builtin_amdgcn_wmma_f32_16x16x4_f32(false, af, false, bf,
                                                       (short)0, cf, false, false);
        }
#pragma unroll
        for (int r = 0; r < 16; ++r) sc[r] = __shfl(cf[r & 7], (r < 8) ? 0 : 16, 32);
#else
        // fallback: per-lane partial dot + shuffle reduction
        for (int r = 0; r < 16; ++r) {
            float p = 0.f;
#pragma unroll
            for (int i = 0; i < 16; ++i) p += tile[r * ND + c0 + i] * smem[LDS_W_OFF / 4 + c0 + i];
            for (int off = 16; off > 0; off >>= 1) p += __shfl_xor(p, off, 32);
            sc[r] = p;
        }
#endif

        // ---- mask + online-softmax rescale ---------------------------------
        bool mk[16];
        float tmax = -2.0f * BIGV;
#pragma unroll
        for (int r = 0; r < 16; ++r) {
            mk[r] = mask[(size_t)b * NS + srow + r] != 0;
            if (!mk[r]) sc[r] = -BIGV;
            cnt += mk[r] ? 1.f : 0.f;
            tmax = fmaxf(tmax, sc[r]);
        }
        const float nm    = fmaxf(m, tmax);
        const float alpha = __expf(m - nm);
        l *= alpha;
#pragma unroll
        for (int i = 0; i < 16; ++i) accv[i] *= alpha;
        m = nm;

        // ---- accumulate stats + softmax numerator (b128 LDS reads) ---------
        for (int r = 0; r < 16; ++r) {
            const float p  = __expf(sc[r] - m);
            const float mf = mk[r] ? 1.f : 0.f;
            l += p;
            const float4* rp = (const float4*)&tile[r * ND + c0];
#pragma unroll
            for (int q = 0; q < 4; ++q) {
                float4 xv = rp[q];
                const int i = q * 4;
                sumv[i+0] += mf * xv.x;  sumv[i+1] += mf * xv.y;
                sumv[i+2] += mf * xv.z;  sumv[i+3] += mf * xv.w;
                if (mk[r]) {
                    maxv[i+0] = fmaxf(maxv[i+0], xv.x); minv[i+0] = fminf(minv[i+0], xv.x);
                    maxv[i+1] = fmaxf(maxv[i+1], xv.y); minv[i+1] = fminf(minv[i+1], xv.y);
                    maxv[i+2] = fmaxf(maxv[i+2], xv.z); minv[i+2] = fminf(minv[i+2], xv.z);
                    maxv[i+3] = fmaxf(maxv[i+3], xv.w); minv[i+3] = fminf(minv[i+3], xv.w);
                }
                accv[i+0] += p * xv.x;  accv[i+1] += p * xv.y;
                accv[i+2] += p * xv.z;  accv[i+3] += p * xv.w;
            }
        }
    }

    // ---- cross-wave merge in LDS, one partial per workgroup ----------------
    float* comb = &smem[LDS_COMB_OFF / 4 + wave * COMB_FLOATS];
#pragma unroll
    for (int i = 0; i < 16; ++i) {
        comb[c0 + i]        = sumv[i];
        comb[512 + c0 + i]  = maxv[i];
        comb[1024 + c0 + i] = minv[i];
        comb[1536 + c0 + i] = accv[i];
    }
    if (lane == 0) { comb[2048] = m; comb[2049] = l; comb[2050] = cnt; }
    __syncthreads();

    if (wave == 0) {
        float Mw[WAVES], Ew[WAVES];
        float Mx = -3.0f * BIGV;
#pragma unroll
        for (int wv = 0; wv < WAVES; ++wv) {
            Mw[wv] = smem[LDS_COMB_OFF / 4 + wv * COMB_FLOATS + 2048];
            Mx = fmaxf(Mx, Mw[wv]);
        }
        float L = 0.f, C = 0.f;
#pragma unroll
        for (int wv = 0; wv < WAVES; ++wv) {
            Ew[wv] = __expf(Mw[wv] - Mx);
            L += smem[LDS_COMB_OFF / 4 + wv * COMB_FLOATS + 2049] * Ew[wv];
            C += smem[LDS_COMB_OFF / 4 + wv * COMB_FLOATS + 2050];
        }
        float* dst = partial + (size_t)blockIdx.x * COMB_FLOATS;
#pragma unroll
        for (int i = 0; i < 16; ++i) {
            const int cidx = c0 + i;
            float s = 0.f, mx = -BIGV, mn = BIGV, ac = 0.f;
#pragma unroll
            for (int wv = 0; wv < WAVES; ++wv) {
                const float* cw = &smem[LDS_COMB_OFF / 4 + wv * COMB_FLOATS];
                s  += cw[cidx];
                mx  = fmaxf(mx, cw[512 + cidx]);
                mn  = fminf(mn, cw[1024 + cidx]);
                ac += cw[1536 + cidx] * Ew[wv];
            }
            dst[cidx]        = s;
            dst[512 + cidx]  = mx;
            dst[1024 + cidx] = mn;
            dst[1536 + cidx] = ac;
        }
        if (lane == 0) { dst[2048] = Mx; dst[2049] = L; dst[2050] = C; }
    }
}

// ---------------------------------------------------------------- pass 2 ----
extern "C" __global__ __launch_bounds__(512)
void mhp_pass2(const float* __restrict__ partial, float* __restrict__ out)
{
    const int b = blockIdx.x;
    const int d = threadIdx.x;

    float Mx = -3.0f * BIGV;
    for (int c = 0; c < NCHUNK; ++c)
        Mx = fmaxf(Mx, partial[((size_t)(b * NCHUNK + c)) * COMB_FLOATS + 2048]);

    float SUM = 0.f, MX = -BIGV, MN = BIGV, ACC = 0.f, L = 0.f, CNT = 0.f;
    for (int c = 0; c < NCHUNK; ++c) {
        const float* p = partial + ((size_t)(b * NCHUNK + c)) * COMB_FLOATS;
        const float ew = __expf(p[2048] - Mx);
        L   += p[2049] * ew;
        CNT += p[2050];
        SUM += p[d];
        MX   = fmaxf(MX, p[512 + d]);
        MN   = fminf(MN, p[1024 + d]);
        ACC += p[1536 + d] * ew;
    }
    float* o = out + (size_t)b * 2048;
    o[d]        = SUM / (CNT + 1e-6f);
    o[512 + d]  = MX;
    o[1024 + d] = MN;
    o[1536 + d] = ACC / L;
}

// ------------------------------------------------------------- launcher ----
extern "C" void kernel_launch(void* const* d_in, const int* in_sizes, int n_in,
                              void* d_out, int out_size, void* d_ws, size_t ws_size,
                              hipStream_t stream)
{
    (void)in_sizes; (void)n_in; (void)out_size; (void)ws_size;
    const float*         x    = (const float*)d_in[0];
    const unsigned char* mask = (const unsigned char*)d_in[1];
    const float*         w    = (const float*)d_in[2];
    float* out  = (float*)d_out;
    float* part = (float*)d_ws;   // needs NB*NCHUNK*COMB_FLOATS*4 = ~4.2 MB

    mhp_pass1<<<dim3(NB * NCHUNK), dim3(WAVES * 32), LDS_BYTES, stream>>>(x, mask, w, part);
    mhp_pass2<<<dim3(NB), dim3(512), 0, stream>>>(part, out);
}